// MEFSSIM_52020643889201
// MI455X (gfx1250) — compile-verified
//
#include <hip/hip_runtime.h>

typedef float v2f __attribute__((ext_vector_type(2)));
typedef float v8f __attribute__((ext_vector_type(8)));

#define IMG 1024
#define TILES_X 64
#define TILES_Y 64
#define NCH 3
#define NTILES (TILES_X * TILES_Y * NCH)
#define NX (NCH * IMG * IMG)
#define NY (6 * NCH * IMG * IMG)
#define PATCH_F (32 * 28)                    // padded patch: 26x26 valid, zero elsewhere
#define TBUF_F (32 * 16)
#define WAVE_F (2 * PATCH_F + 3 * TBUF_F)    // 3328 floats per wave

__device__ __forceinline__ v8f wmma4(v2f a, v2f b, v8f c) {
  // D = A(16x4,f32) * B(4x16,f32) + C(16x16,f32)
  return __builtin_amdgcn_wmma_f32_16x16x4_f32(false, a, false, b, (short)0, c,
                                               false, false);
}

__device__ __forceinline__ v8f vzero() {
  v8f z = {0.f, 0.f, 0.f, 0.f, 0.f, 0.f, 0.f, 0.f};
  return z;
}

// One-time zero fill of pad rows 26..31 (later fills only rewrite rows 0..25).
__device__ __forceinline__ void zero_pad_rows(float* dst, int lane) {
  if (lane < 28) {
#pragma clang loop unroll(disable)
    for (int r = 26; r < 32; ++r) dst[r * 28 + lane] = 0.0f;
  }
}

// Fill rows 0..25 (cols 26..27 get zeros via the column predicate).
// lane = column -> coalesced 26-row global streams. Keep the loop ROLLED:
// unrolling bloats static code ~26x for zero benefit (latency is hidden by
// prefetch + the WMMA chain).
__device__ void fill_patch(const float* __restrict__ src, float* dst, int r0,
                           int c0, float s, int lane) {
  const int c = lane;
  const int gc = c0 + c - 5;
  const bool cok = (c < 26) & ((unsigned)gc < (unsigned)IMG);
  const bool cst = (c < 28);
#pragma clang loop unroll(disable)
  for (int r = 0; r < 26; ++r) {
    const int gr = r0 + r - 5;
    float v = 0.0f;
    if (cok && ((unsigned)gr < (unsigned)IMG)) v = src[gr * IMG + gc] * s;
    if (cst) dst[r * 28 + c] = v;
  }
}

// Prefetch one patch row base per lane for the next exposure (gfx1250
// global_prefetch path; whole cacheline pulled per address, no counters).
__device__ __forceinline__ void prefetch_patch(const float* __restrict__ src,
                                               int r0, int c0, int lane) {
  const int gr = r0 + lane - 5;
  const int gc = min(max(c0 - 5, 0), IMG - 1);
  if (lane < 26 && (unsigned)gr < (unsigned)IMG)
    __builtin_prefetch(src + gr * IMG + gc, 0, 1);
}

// ---------------- scale (faithful 255-vs-1 branch) ----------------
__device__ __forceinline__ unsigned fmap(float f) {
  unsigned b = __float_as_uint(f);
  return (b & 0x80000000u) ? ~b : (b | 0x80000000u);  // monotonic map
}

__global__ void mef_init(unsigned* u) {
  u[0] = 0u;
  u[1] = 0u;
}

__global__ void mef_max(const float* __restrict__ X,
                        const float* __restrict__ Ys, unsigned* u) {
  unsigned mx = 0u, my = 0u;
  int g = blockIdx.x * blockDim.x + threadIdx.x;
  int stride = gridDim.x * blockDim.x;
#pragma clang loop unroll(disable)
  for (int i = g; i < NX; i += stride) mx = max(mx, fmap(X[i]));
#pragma clang loop unroll(disable)
  for (int i = g; i < NY; i += stride) my = max(my, fmap(Ys[i]));
#pragma unroll
  for (int off = 16; off > 0; off >>= 1) {
    mx = max(mx, (unsigned)__shfl_xor((int)mx, off, 32));
    my = max(my, (unsigned)__shfl_xor((int)my, off, 32));
  }
  __shared__ unsigned smx[8], smy[8];
  int lane = threadIdx.x & 31, wid = threadIdx.x >> 5;
  if (lane == 0) { smx[wid] = mx; smy[wid] = my; }
  __syncthreads();
  if (threadIdx.x == 0) {
    for (int w = 1; w < 8; ++w) { mx = max(mx, smx[w]); my = max(my, smy[w]); }
    atomicMax(&u[0], mx);
    atomicMax(&u[1], my);
  }
}

__global__ void mef_scale(const unsigned* u, float* wsf) {
  unsigned a = u[0], b = u[1];
  unsigned ba = (a & 0x80000000u) ? (a ^ 0x80000000u) : ~a;
  unsigned bb = (b & 0x80000000u) ? (b ^ 0x80000000u) : ~b;
  float fx = __uint_as_float(ba);
  float fy = __uint_as_float(bb);
  wsf[2] = (fx <= 1.0f && fy <= 1.0f) ? 255.0f : 1.0f;
}

// ---------------- fused MEF-SSIM main kernel ----------------
// Separable 11-tap Gaussian conv per 16x16 tile via V_WMMA_F32_16X16X4_F32:
//   Horizontal: T(26x16) = Patch(26x28) x Wh(28x16)   [2 row-groups x 7 wmma]
//   Vertical:   D(16x16) = Wv(16x28) x T(28x16)       [7 wmma]
// Per-lane banded-Gaussian operand g[k - l15] is identical for Wh-as-B and
// Wv-as-A -> precomputed once into 7 v2f registers (wop).
__global__ void __launch_bounds__(128) mef_main(
    const float* __restrict__ X, const float* __restrict__ Ys,
    const float* __restrict__ win, const float* __restrict__ wsf,
    float* __restrict__ partials) {
  __shared__ float smem[4 * WAVE_F + 48];
  float* wext = smem + 4 * WAVE_F;  // zero-padded 1D gaussian, index j+15, j in [-15,27]
  const int tid = threadIdx.x;
  if (tid < 43) {
    float v = 0.0f;
    if (tid >= 15 && tid <= 25) {  // g[i] = sum_j window[i][j]
      const int i = tid - 15;
#pragma clang loop unroll(disable)
      for (int j = 0; j < 11; ++j) v += win[i * 11 + j];
    }
    wext[tid] = v;
  }
  __syncthreads();

  const int wid = tid >> 5, lane = tid & 31;
  const int half = lane >> 4, l15 = lane & 15;
  float* xp = smem + wid * WAVE_F;
  float* yp = xp + PATCH_F;
  float* Tb0 = yp + PATCH_F;
  float* Tb1 = Tb0 + TBUF_F;
  float* Tb2 = Tb1 + TBUF_F;

  const int tx = blockIdx.x * 4 + wid;
  const int ty = blockIdx.y;
  const int ch = blockIdx.z;
  const int r0 = ty * 16, c0 = tx * 16;
  const float s = wsf[2];
  const float* __restrict__ Ybase = Ys + ch * IMG * IMG;

  // Loop-invariant Gaussian WMMA operand (horizontal-B == vertical-A values)
  v2f wop[7];
#pragma unroll
  for (int kc = 0; kc < 7; ++kc) {
    const int ka = kc * 4 + half * 2;
    wop[kc].x = wext[ka - l15 + 15];
    wop[kc].y = wext[ka + 1 - l15 + 15];
  }

  zero_pad_rows(xp, lane);
  zero_pad_rows(yp, lane);
  fill_patch(X + ch * IMG * IMG, xp, r0, c0, s, lane);
  prefetch_patch(Ybase, r0, c0, lane);  // k=0 exposure

  v2f xa[14];  // per-lane A-operand patch values, resident for whole kernel
#pragma unroll
  for (int rg = 0; rg < 2; ++rg)
#pragma unroll
    for (int kc = 0; kc < 7; ++kc)
      xa[rg * 7 + kc] =
          *(const v2f*)(xp + (l15 + rg * 16) * 28 + kc * 4 + half * 2);

  // ---- conv(X), conv(X^2): two interleaved WMMA chains ----
#pragma unroll
  for (int rg = 0; rg < 2; ++rg) {
    v8f a0 = vzero(), a1 = vzero();
#pragma unroll
    for (int kc = 0; kc < 7; ++kc) {
      v2f x = xa[rg * 7 + kc];
      v2f xx;
      xx.x = x.x * x.x;
      xx.y = x.y * x.y;
      a0 = wmma4(x, wop[kc], a0);
      a1 = wmma4(xx, wop[kc], a1);
    }
    const int br = rg * 16 + half * 8;
#pragma unroll
    for (int v = 0; v < 8; ++v) {
      Tb0[(br + v) * 16 + l15] = a0[v];
      Tb1[(br + v) * 16 + l15] = a1[v];
    }
  }
  v8f muX = vzero(), exx = vzero();
#pragma unroll
  for (int kc = 0; kc < 7; ++kc) {
    const int kb = kc * 4 + half * 2;
    v2f b0, b1;
    b0.x = Tb0[kb * 16 + l15];
    b0.y = Tb0[(kb + 1) * 16 + l15];
    b1.x = Tb1[kb * 16 + l15];
    b1.y = Tb1[(kb + 1) * 16 + l15];
    muX = wmma4(wop[kc], b0, muX);
    exx = wmma4(wop[kc], b1, exx);
  }
  v8f sigX;
#pragma unroll
  for (int i = 0; i < 8; ++i) sigX[i] = exx[i] - muX[i] * muX[i];

  v8f bSig = vzero(), bCs = vzero();
  const float C2 = 0.0009f, EPS = 1e-8f;

  for (int k = 0; k < 6; ++k) {
    fill_patch(Ybase + k * NCH * IMG * IMG, yp, r0, c0, s, lane);
    if (k < 5)  // hide next exposure's HBM latency behind this k's 63 WMMAs
      prefetch_patch(Ybase + (k + 1) * NCH * IMG * IMG, r0, c0, lane);

    v2f ya[14];
#pragma unroll
    for (int rg = 0; rg < 2; ++rg)
#pragma unroll
      for (int kc = 0; kc < 7; ++kc)
        ya[rg * 7 + kc] =
            *(const v2f*)(yp + (l15 + rg * 16) * 28 + kc * 4 + half * 2);

    // ---- conv(Y), conv(Y^2), conv(X*Y): three interleaved WMMA chains ----
#pragma unroll
    for (int rg = 0; rg < 2; ++rg) {
      v8f a0 = vzero(), a1 = vzero(), a2 = vzero();
#pragma unroll
      for (int kc = 0; kc < 7; ++kc) {
        v2f x = xa[rg * 7 + kc];
        v2f y = ya[rg * 7 + kc];
        v2f yy, xy;
        yy.x = y.x * y.x;
        yy.y = y.y * y.y;
        xy.x = x.x * y.x;
        xy.y = x.y * y.y;
        a0 = wmma4(y, wop[kc], a0);
        a1 = wmma4(yy, wop[kc], a1);
        a2 = wmma4(xy, wop[kc], a2);
      }
      const int br = rg * 16 + half * 8;
#pragma unroll
      for (int v = 0; v < 8; ++v) {
        Tb0[(br + v) * 16 + l15] = a0[v];
        Tb1[(br + v) * 16 + l15] = a1[v];
        Tb2[(br + v) * 16 + l15] = a2[v];
      }
    }
    v8f muY = vzero(), eyy = vzero(), exy = vzero();
#pragma unroll
    for (int kc = 0; kc < 7; ++kc) {
      const int kb = kc * 4 + half * 2;
      v2f b0, b1, b2;
      b0.x = Tb0[kb * 16 + l15];
      b0.y = Tb0[(kb + 1) * 16 + l15];
      b1.x = Tb1[kb * 16 + l15];
      b1.y = Tb1[(kb + 1) * 16 + l15];
      b2.x = Tb2[kb * 16 + l15];
      b2.y = Tb2[(kb + 1) * 16 + l15];
      muY = wmma4(wop[kc], b0, muY);
      eyy = wmma4(wop[kc], b1, eyy);
      exy = wmma4(wop[kc], b2, exy);
    }

#pragma unroll
    for (int i = 0; i < 8; ++i) {
      float sy = eyy[i] - muY[i] * muY[i];
      float sxy = exy[i] - muX[i] * muY[i];
      float cs = (2.0f * sxy + C2) / (sigX[i] + sy + C2 + EPS);
      bool cond = (k == 0) || (sy > bSig[i]);  // strict > keeps first argmax
      bSig[i] = cond ? sy : bSig[i];
      bCs[i] = cond ? cs : bCs[i];
    }
  }

  float t = 0.0f;
#pragma unroll
  for (int i = 0; i < 8; ++i) t += bCs[i];
#pragma unroll
  for (int off = 16; off > 0; off >>= 1) t += __shfl_xor(t, off, 32);
  if (lane == 0) partials[ch * (TILES_X * TILES_Y) + ty * TILES_X + tx] = t;
}

__global__ void mef_final(const float* __restrict__ partials,
                          float* __restrict__ out) {
  __shared__ float sm[256];
  float s = 0.0f;
#pragma clang loop unroll(disable)
  for (int i = threadIdx.x; i < NTILES; i += 256) s += partials[i];  // fixed order
  sm[threadIdx.x] = s;
  __syncthreads();
  for (int off = 128; off > 0; off >>= 1) {
    if (threadIdx.x < off) sm[threadIdx.x] += sm[threadIdx.x + off];
    __syncthreads();
  }
  if (threadIdx.x == 0) out[0] = sm[0] / (float)(NCH * IMG * IMG);
}

extern "C" void kernel_launch(void* const* d_in, const int* in_sizes, int n_in,
                              void* d_out, int out_size, void* d_ws,
                              size_t ws_size, hipStream_t stream) {
  const float* X = (const float*)d_in[0];    // (1,3,1024,1024)
  const float* Ys = (const float*)d_in[1];   // (6,3,1024,1024)
  const float* win = (const float*)d_in[2];  // (3,1,11,11)
  float* out = (float*)d_out;
  float* wsf = (float*)d_ws;
  unsigned* wsu = (unsigned*)d_ws;
  float* partials = wsf + 4;  // NTILES floats

  mef_init<<<1, 1, 0, stream>>>(wsu);
  mef_max<<<1024, 256, 0, stream>>>(X, Ys, wsu);
  mef_scale<<<1, 1, 0, stream>>>(wsu, wsf);
  mef_main<<<dim3(TILES_X / 4, TILES_Y, NCH), 128, 0, stream>>>(X, Ys, win, wsf,
                                                                partials);
  mef_final<<<1, 256, 0, stream>>>(partials, out);
}